// FullDenseRetrievalLoss_63702954934991
// MI455X (gfx1250) — compile-verified
//
#include <hip/hip_runtime.h>
#include <math.h>

typedef __attribute__((ext_vector_type(2))) float v2f;
typedef __attribute__((ext_vector_type(8))) float v8f;

namespace {
constexpr int N = 4096;
constexpr int V = 65536;
constexpr int D = 256;
constexpr float TAU = 0.1f;
constexpr int QS_STRIDE = 260;          // 16 rows * 260 floats, bank-conflict-free
constexpr int WAVES = 8;
constexpr int COLS_PER_ITER = WAVES * 16;   // 128 columns per workgroup iteration
constexpr int ITERS = V / COLS_PER_ITER;    // 512
}

// One wave per row: compute 1/max(||row||, eps), with 1/TAU folded in for queries.
__global__ __launch_bounds__(256) void rownorm_kernel(const float* __restrict__ q,
                                                      const float* __restrict__ k,
                                                      float* __restrict__ qinv,
                                                      float* __restrict__ kinv) {
  int gwave = (int)((blockIdx.x * blockDim.x + threadIdx.x) >> 5);
  int lane = threadIdx.x & 31;
  const float* src;
  float* dst;
  float scale;
  if (gwave < N) {
    src = q + (size_t)gwave * D; dst = qinv + gwave; scale = 1.0f / TAU;
  } else if (gwave < N + V) {
    int r = gwave - N;
    src = k + (size_t)r * D; dst = kinv + r; scale = 1.0f;
  } else {
    return;
  }
  float4 x0 = *(const float4*)(src + lane * 8);
  float4 x1 = *(const float4*)(src + lane * 8 + 4);
  float acc = x0.x * x0.x + x0.y * x0.y + x0.z * x0.z + x0.w * x0.w +
              x1.x * x1.x + x1.y * x1.y + x1.z * x1.z + x1.w * x1.w;
#pragma unroll
  for (int m = 1; m <= 16; m <<= 1) acc += __shfl_xor(acc, m, 32);
  if (lane == 0) {
    float n = sqrtf(acc);
    *dst = scale / fmaxf(n, 1e-12f);
  }
}

// One block = 16 query rows.  8 waves each own a distinct 16-col key tile per
// iteration; 64 x V_WMMA_F32_16X16X4_F32 per tile; flash-softmax accumulation.
__global__ __launch_bounds__(256) void FullDenseRetrievalLoss_main(
    const float* __restrict__ q, const float* __restrict__ kmat,
    const int* __restrict__ labels, const float* __restrict__ qinv,
    const float* __restrict__ kinv, float* __restrict__ nll) {
  __shared__ float qs[16 * QS_STRIDE];
  __shared__ float wm[WAVES][16];
  __shared__ float wssum[WAVES][16];
  __shared__ float lblv[16];
  __shared__ int lab[16];

  const int tid = threadIdx.x;
  const int wave = tid >> 5;
  const int lane = tid & 31;
  const int lane15 = lane & 15;
  const int hi = lane >> 4;              // 0: K 0..1 / rows 0..7, 1: K 2..3 / rows 8..15
  const int row_base = blockIdx.x * 16;

  // Stage queries scaled by 1/(||q||*TAU) into LDS (coalesced, padded stride).
  for (int idx = tid; idx < 16 * D; idx += 256) {
    int r = idx >> 8;               // D == 256
    int d = idx & (D - 1);
    qs[r * QS_STRIDE + d] = q[(size_t)(row_base + r) * D + d] * qinv[row_base + r];
  }
  if (tid < 16) {
    lab[tid] = labels[row_base + tid];
    lblv[tid] = 0.0f;
  }
  __syncthreads();

  // A-fragment source: lane L holds Q[row=L&15][k_base + (L>=16?2:0) + {0,1}]
  const float* arow = qs + lane15 * QS_STRIDE + hi * 2;

  float m[8], s[8];
#pragma unroll
  for (int r = 0; r < 8; ++r) { m[r] = -1e30f; s[r] = 0.0f; }

  for (int it = 0; it < ITERS; ++it) {
    const int col_base = it * COLS_PER_ITER + wave * 16;
    const int mycol = col_base + lane15;
    const float rnk = kinv[mycol];  // per-lane: all this lane's B elements share one key row
    const float* krow = kmat + (size_t)mycol * D + hi * 2;

    if (it + 1 < ITERS)
      __builtin_prefetch(krow + (size_t)COLS_PER_ITER * D, 0, 1);

    v8f acc = {};
#pragma unroll 8
    for (int kk = 0; kk < D / 4; ++kk) {
      v2f a = *(const v2f*)(arow + kk * 4);   // ds_load_b64
      v2f b = *(const v2f*)(krow + kk * 4);   // global_load_b64, L2-resident stream
      b *= rnk;                               // fold key normalization into B fragment
      acc = __builtin_amdgcn_wmma_f32_16x16x4_f32(
          false, a, false, b, (short)0, acc, false, false);
    }

    // Online softmax: each lane owns 8 (row, col-subset) partials.
#pragma unroll
    for (int r = 0; r < 8; ++r) {
      float x = acc[r];
      float mn = fmaxf(m[r], x);
      s[r] = s[r] * __expf(m[r] - mn) + __expf(x - mn);
      m[r] = mn;
    }
    // Capture the logit at the label column (exactly one writer per row).
#pragma unroll
    for (int r = 0; r < 8; ++r) {
      int rowl = (hi << 3) + r;
      if (lab[rowl] == mycol) lblv[rowl] = acc[r];
    }
  }

  // Merge (m,s) across the 16 lanes of each half-wave (rows stay within a half).
#pragma unroll
  for (int mask = 1; mask <= 8; mask <<= 1) {
#pragma unroll
    for (int r = 0; r < 8; ++r) {
      float mo = __shfl_xor(m[r], mask, 32);
      float so = __shfl_xor(s[r], mask, 32);
      float mn = fmaxf(m[r], mo);
      s[r] = s[r] * __expf(m[r] - mn) + so * __expf(mo - mn);
      m[r] = mn;
    }
  }
  if (lane15 == 0) {
#pragma unroll
    for (int r = 0; r < 8; ++r) {
      wm[wave][(hi << 3) + r] = m[r];
      wssum[wave][(hi << 3) + r] = s[r];
    }
  }
  __syncthreads();

  // Merge the 8 waves' partials, finish: nll = m + log(s) - logit_label.
  if (tid < 16) {
    float mm = -1e30f, ss = 0.0f;
#pragma unroll
    for (int w = 0; w < WAVES; ++w) {
      float mo = wm[w][tid], so = wssum[w][tid];
      float mn = fmaxf(mm, mo);
      ss = ss * __expf(mm - mn) + so * __expf(mo - mn);
      mm = mn;
    }
    nll[row_base + tid] = mm + __logf(ss) - lblv[tid];
  }
}

// Deterministic mean over the 4096 per-row NLLs.
__global__ __launch_bounds__(256) void reduce_kernel(const float* __restrict__ nll,
                                                     float* __restrict__ out) {
  __shared__ float sh[256];
  float a = 0.0f;
  for (int i = threadIdx.x; i < N; i += 256) a += nll[i];
  sh[threadIdx.x] = a;
  __syncthreads();
  for (int s = 128; s > 0; s >>= 1) {
    if ((int)threadIdx.x < s) sh[threadIdx.x] += sh[threadIdx.x + s];
    __syncthreads();
  }
  if (threadIdx.x == 0) out[0] = sh[0] / (float)N;
}

extern "C" void kernel_launch(void* const* d_in, const int* in_sizes, int n_in,
                              void* d_out, int out_size, void* d_ws, size_t ws_size,
                              hipStream_t stream) {
  const float* q = (const float*)d_in[0];
  const float* k = (const float*)d_in[1];
  const int* labels = (const int*)d_in[2];

  float* qinv = (float*)d_ws;       // N floats
  float* kinv = qinv + N;           // V floats
  float* nll = kinv + V;            // N floats   (total ~288 KB of workspace)

  int total_rows = N + V;           // one wave per row, 8 waves per block
  int norm_blocks = (total_rows + 7) / 8;
  rownorm_kernel<<<norm_blocks, 256, 0, stream>>>(q, k, qinv, kinv);

  FullDenseRetrievalLoss_main<<<N / 16, 256, 0, stream>>>(q, k, labels, qinv,
                                                          kinv, nll);

  reduce_kernel<<<1, 256, 0, stream>>>(nll, (float*)d_out);
}